// S2V_57595511439900
// MI455X (gfx1250) — compile-verified
//
#include <hip/hip_runtime.h>

// S2V-style GNN layer for MI455X (gfx1250, wave32).
// Pipeline: zero -> coeff -> h_mu(WMMA f32) -> edge scatter(atomics) -> final(WMMA f32 + epilogue)

typedef __attribute__((ext_vector_type(2))) float v2f;
typedef __attribute__((ext_vector_type(8))) float v8f;

__device__ __forceinline__ v8f wmma_f32_4(v2f a, v2f b, v8f c) {
    // V_WMMA_F32_16X16X4_F32 : D(16x16 f32) = A(16x4 f32) * B(4x16 f32) + C
    return __builtin_amdgcn_wmma_f32_16x16x4_f32(
        /*neg_a=*/false, a, /*neg_b=*/false, b,
        /*c_mod=*/(short)0, c, /*reuse_a=*/false, /*reuse_b=*/false);
}

// B staged in LDS as interleaved K-pairs: Blds[(k>>1)*160 + o*2 + (k&1)] = B[k,o].
// -> B fragment {B[kb,o], B[kb+1,o]} is one aligned ds_load_b64.
// Row stride 160 words == 32 (mod 64 banks): hi/lo half-waves hit disjoint bank halves.
#define BSTRIDE 160
#define BWORDS  (32 * BSTRIDE)

// ---------------------------------------------------------------------------
__global__ void k_zero(float* p, int n) {
    int i = blockIdx.x * blockDim.x + threadIdx.x;
    if (i < n) p[i] = 0.0f;
}

// C[0..63]=C1p, [64..127]=C1m, [128..191]=C2p, [192..255]=C2m
// C1p[o] = sum_k W4[o,k]*relu(W1[k]) etc.  (rank-4 collapse of the h_x / h_e scatter)
__global__ void k_coeff(const float* __restrict__ W1, const float* __restrict__ W2,
                        const float* __restrict__ W4, float* __restrict__ C) {
    int o = threadIdx.x;  // 64 threads
    float c1p = 0.f, c1m = 0.f, c2p = 0.f, c2m = 0.f;
    for (int k = 0; k < 64; ++k) {
        float w4a = W4[o * 192 + k];
        float w4b = W4[o * 192 + 64 + k];
        c1p += w4a * fmaxf(W1[k], 0.f);
        c1m += w4a * fmaxf(-W1[k], 0.f);
        c2p += w4b * fmaxf(W2[k], 0.f);
        c2m += w4b * fmaxf(-W2[k], 0.f);
    }
    C[o] = c1p; C[64 + o] = c1m; C[128 + o] = c2p; C[192 + o] = c2m;
}

// ---------------------------------------------------------------------------
// h_mu = relu(mu @ W3^T), [N,64] x [64,64] via V_WMMA_F32_16X16X4_F32.
// One wave computes a 16-row x 64-col tile (4 accumulators, 16 K-steps of 4).
__global__ void k_hmu(const float* __restrict__ mu, const float* __restrict__ W3,
                      float* __restrict__ h_mu, int n_tiles) {
    __shared__ float Blds[BWORDS];
    for (int idx = threadIdx.x; idx < 64 * 64; idx += blockDim.x) {
        int o = idx >> 6, k = idx & 63;
        Blds[(k >> 1) * BSTRIDE + o * 2 + (k & 1)] = W3[idx];  // B[k,o] = W3[o,k]
    }
    __syncthreads();

    const int lane = threadIdx.x & 31;
    const int lo = lane & 15, hi = lane >> 4;
    const int wpb = blockDim.x >> 5;
    const int gwave = blockIdx.x * wpb + (threadIdx.x >> 5);
    const int nwaves = gridDim.x * wpb;

    for (int t = gwave; t < n_tiles; t += nwaves) {
        const int base = t * 16;
        // A-frag: lane holds row (base+lo), K = kk*4 + hi*2 (+0,+1)  (8B-aligned pair)
        const float* arow = mu + (base + lo) * 64 + hi * 2;
        v8f acc0 = {}, acc1 = {}, acc2 = {}, acc3 = {};
#pragma unroll 4
        for (int kk = 0; kk < 16; ++kk) {
            v2f a = *(const v2f*)(arow + kk * 4);
            // kb = kk*4 + hi*2 (even) -> kpair row = kk*2 + hi
            const float* bp = Blds + (kk * 2 + hi) * BSTRIDE + lo * 2;
            acc0 = wmma_f32_4(a, *(const v2f*)(bp +  0), acc0);
            acc1 = wmma_f32_4(a, *(const v2f*)(bp + 32), acc1);
            acc2 = wmma_f32_4(a, *(const v2f*)(bp + 64), acc2);
            acc3 = wmma_f32_4(a, *(const v2f*)(bp + 96), acc3);
        }
        // D layout: VGPR r -> rows (base+r) [lanes 0-15] / (base+r+8) [lanes 16-31], col=lane%16
#pragma unroll
        for (int r = 0; r < 8; ++r) {
            int row = base + r + hi * 8;
            float* o = h_mu + row * 64 + lo;
            o[0]  = fmaxf(acc0[r], 0.f);
            o[16] = fmaxf(acc1[r], 0.f);
            o[32] = fmaxf(acc2[r], 0.f);
            o[48] = fmaxf(acc3[r], 0.f);
        }
    }
}

// ---------------------------------------------------------------------------
// One wave32 per edge: lanes scatter the 64-float h_mu[dst] row into aggmu[src]
// (coalesced f32 atomics, L2-resident); lanes 0-3 each do one scalar-sum atomic.
__global__ void k_edge(const int* __restrict__ ei, const float* __restrict__ ew,
                       const float* __restrict__ x, const float* __restrict__ h_mu,
                       float* __restrict__ aggmu, float* __restrict__ S,
                       int N, int E) {
    const int lane = threadIdx.x & 31;
    const int e = (blockIdx.x * blockDim.x + threadIdx.x) >> 5;
    if (e >= E) return;
    const int s = __builtin_nontemporal_load(&ei[e]);      // scatter row (src)
    const int d = __builtin_nontemporal_load(&ei[E + e]);  // gather row  (dst)
    const float* hm = h_mu + d * 64;
    float* ag = aggmu + s * 64;
    atomicAdd(&ag[lane],      hm[lane]);
    atomicAdd(&ag[lane + 32], hm[lane + 32]);
    if (lane < 4) {
        const float xd = x[d];
        const float w  = __builtin_nontemporal_load(&ew[e]);
        const float v = (lane == 0) ? fmaxf(xd, 0.f)
                      : (lane == 1) ? fmaxf(-xd, 0.f)
                      : (lane == 2) ? fmaxf(w, 0.f)
                                    : fmaxf(-w, 0.f);
        atomicAdd(&S[lane * N + s], v);
    }
}

// ---------------------------------------------------------------------------
// out = relu( relu(x*W1) + h_mu + relu( aggmu @ W4[:,128:192]^T + rank-4 term ) )
__global__ void k_final(const float* __restrict__ aggmu, const float* __restrict__ W4,
                        const float* __restrict__ C, const float* __restrict__ x,
                        const float* __restrict__ W1, const float* __restrict__ h_mu,
                        const float* __restrict__ S, float* __restrict__ out,
                        int N, int n_tiles) {
    __shared__ float Blds[BWORDS];  // B[k,o] = W4[o, 128+k], interleaved K-pairs
    __shared__ float Cl[256];
    __shared__ float W1l[64];
    for (int idx = threadIdx.x; idx < 64 * 64; idx += blockDim.x) {
        int o = idx >> 6, k = idx & 63;
        Blds[(k >> 1) * BSTRIDE + o * 2 + (k & 1)] = W4[o * 192 + 128 + k];
    }
    for (int idx = threadIdx.x; idx < 256; idx += blockDim.x) Cl[idx] = C[idx];
    if (threadIdx.x < 64) W1l[threadIdx.x] = W1[threadIdx.x];
    __syncthreads();

    const int lane = threadIdx.x & 31;
    const int lo = lane & 15, hi = lane >> 4;
    const int wpb = blockDim.x >> 5;
    const int gwave = blockIdx.x * wpb + (threadIdx.x >> 5);
    const int nwaves = gridDim.x * wpb;

    for (int t = gwave; t < n_tiles; t += nwaves) {
        const int base = t * 16;
        const float* arow = aggmu + (base + lo) * 64 + hi * 2;
        v8f acc0 = {}, acc1 = {}, acc2 = {}, acc3 = {};
#pragma unroll 4
        for (int kk = 0; kk < 16; ++kk) {
            v2f a = *(const v2f*)(arow + kk * 4);
            const float* bp = Blds + (kk * 2 + hi) * BSTRIDE + lo * 2;
            acc0 = wmma_f32_4(a, *(const v2f*)(bp +  0), acc0);
            acc1 = wmma_f32_4(a, *(const v2f*)(bp + 32), acc1);
            acc2 = wmma_f32_4(a, *(const v2f*)(bp + 64), acc2);
            acc3 = wmma_f32_4(a, *(const v2f*)(bp + 96), acc3);
        }
#pragma unroll
        for (int r = 0; r < 8; ++r) {
            const int row = base + r + hi * 8;
            const float sxp = S[row], sxm = S[N + row];
            const float sep = S[2 * N + row], sem = S[3 * N + row];
            const float xr = x[row];
            const float* hm = h_mu + row * 64 + lo;
            float* orow = out + row * 64 + lo;
#pragma unroll
            for (int ot = 0; ot < 4; ++ot) {
                const int col = ot * 16 + lo;
                float accv = (ot == 0) ? acc0[r] : (ot == 1) ? acc1[r]
                           : (ot == 2) ? acc2[r] : acc3[r];
                float aggv = accv + sxp * Cl[col] + sxm * Cl[64 + col]
                                  + sep * Cl[128 + col] + sem * Cl[192 + col];
                float res = fmaxf(xr * W1l[col], 0.f) + hm[ot * 16] + fmaxf(aggv, 0.f);
                orow[ot * 16] = fmaxf(res, 0.f);
            }
        }
    }
}

// ---------------------------------------------------------------------------
extern "C" void kernel_launch(void* const* d_in, const int* in_sizes, int n_in,
                              void* d_out, int out_size, void* d_ws, size_t ws_size,
                              hipStream_t stream) {
    const float* mu = (const float*)d_in[0];   // [N,64]
    const float* x  = (const float*)d_in[1];   // [N,1]
    const int*   ei = (const int*)d_in[2];     // [2,E] (integer inputs arrive as int32)
    const float* ew = (const float*)d_in[3];   // [E]
    const float* W1 = (const float*)d_in[4];   // [64,1]
    const float* W2 = (const float*)d_in[5];   // [64,1]
    const float* W3 = (const float*)d_in[6];   // [64,64]
    const float* W4 = (const float*)d_in[7];   // [64,192]
    float* out = (float*)d_out;

    const int N = in_sizes[1];  // 50000
    const int E = in_sizes[3];  // 1600000

    float* ws    = (float*)d_ws;
    float* h_mu  = ws;                        // N*64
    float* aggmu = ws + (size_t)N * 64;       // N*64
    float* S     = ws + (size_t)N * 128;      // 4*N  (Sxp,Sxm,Sep,Sem)
    float* C     = ws + (size_t)N * 132;      // 256

    const int nz = N * 68;  // aggmu + S (atomically accumulated -> must start at 0)
    k_zero<<<dim3((nz + 255) / 256), dim3(256), 0, stream>>>(aggmu, nz);
    k_coeff<<<dim3(1), dim3(64), 0, stream>>>(W1, W2, W4, C);

    const int n_tiles = N / 16;                       // 3125 (N divisible by 16)
    const int gblocks = (n_tiles + 7) / 8;            // 8 waves/block, 1 tile/wave
    k_hmu<<<dim3(gblocks), dim3(256), 0, stream>>>(mu, W3, h_mu, n_tiles);

    const int eblocks = (E + 7) / 8;                  // 1 wave32 per edge
    k_edge<<<dim3(eblocks), dim3(256), 0, stream>>>(ei, ew, x, h_mu, aggmu, S, N, E);

    k_final<<<dim3(gblocks), dim3(256), 0, stream>>>(aggmu, W4, C, x, W1, h_mu, S, out,
                                                     N, n_tiles);
}